// TemporalConsistencyAttention_57947698758216
// MI455X (gfx1250) — compile-verified
//
#include <hip/hip_runtime.h>
#include <hip/hip_bf16.h>
#include <math.h>

// ---------------------------------------------------------------------------
// TemporalConsistencyAttention for MI455X (gfx1250, wave32)
// M=3 modes, B=64, D=512, HIST=5, SMOOTH=0.8
// ---------------------------------------------------------------------------

typedef __attribute__((ext_vector_type(2))) float v2f;
typedef __attribute__((ext_vector_type(8))) float v8f;

#define MM 3
#define BB 64
#define DD 512
#define SMOOTH 0.8f
#define INV_SCALE 0.04419417382415922f   // 1/sqrt(512)

// ---------------- wave32 / block reductions --------------------------------
__device__ __forceinline__ float wave_sum(float v) {
#pragma unroll
  for (int o = 16; o > 0; o >>= 1) v += __shfl_down(v, o, 32);
  return v;
}
__device__ __forceinline__ float wave_max(float v) {
#pragma unroll
  for (int o = 16; o > 0; o >>= 1) v = fmaxf(v, __shfl_down(v, o, 32));
  return v;
}
// blockDim.x == 256 (8 waves). s must have >= 8 floats.
__device__ __forceinline__ float block_sum(float v, float* s) {
  const int lane = threadIdx.x & 31, w = threadIdx.x >> 5;
  v = wave_sum(v);
  __syncthreads();
  if (lane == 0) s[w] = v;
  __syncthreads();
  if (w == 0) {
    float x = (lane < 8) ? s[lane] : 0.0f;
    x = wave_sum(x);
    if (lane == 0) s[0] = x;
  }
  __syncthreads();
  return s[0];
}
__device__ __forceinline__ float block_max(float v, float* s) {
  const int lane = threadIdx.x & 31, w = threadIdx.x >> 5;
  v = wave_max(v);
  __syncthreads();
  if (lane == 0) s[w] = v;
  __syncthreads();
  if (w == 0) {
    float x = (lane < 8) ? s[lane] : -3.4e38f;
    x = wave_max(x);
    if (lane == 0) s[0] = x;
  }
  __syncthreads();
  return s[0];
}
__device__ __forceinline__ float sigmoidf(float x) {
  return 1.0f / (1.0f + __expf(-x));
}

// ---------------- fp32 WMMA GEMM:  Y[m,n] = sum_k A[m,k]*Wt[n,k] (+epilogue)
// One wave per 16x16 output tile; V_WMMA_F32_16X16X4_F32, K-step 4.
// A layout (ISA 7.12.2): lane<16 -> row=lane, K=k0..k0+1; lane>=16 -> K=k0+2..3
// B mirrored with lane = output column.  C/D: vgpr v -> M=v (+8 lanes>=16),
// N = lane&15.
// MODE 0: +bias            (projection pre-LN)
// MODE 1: relu(+bias)      (importance hidden)
// MODE 2: extra * sigmoid(+bias)   (wf = proj * channel weight)
template <int MODE>
__global__ void gemm_wmma(const float* __restrict__ A,
                          const float* __restrict__ Wt,
                          const float* __restrict__ bias,
                          const float* __restrict__ extra,
                          float* __restrict__ out, int K, int Ncols) {
  const int mode = blockIdx.z;
  A    += mode * BB * K;
  Wt   += mode * Ncols * K;
  bias += mode * Ncols;
  out  += mode * BB * Ncols;
  if (MODE == 2) extra += mode * BB * Ncols;

  const int nbase = blockIdx.x * 16;
  const int mbase = blockIdx.y * 16;
  const int lane  = threadIdx.x;        // 0..31
  const int l16   = lane & 15;
  const int khalf = (lane >> 4) << 1;   // 0 or 2

  const float* ap = A  + (mbase + l16) * K + khalf;
  const float* bp = Wt + (nbase + l16) * K + khalf;

  v8f acc = {};
  for (int k0 = 0; k0 < K; k0 += 4) {
    v2f av = *(const v2f*)(ap + k0);
    v2f bv = *(const v2f*)(bp + k0);
    acc = __builtin_amdgcn_wmma_f32_16x16x4_f32(
        /*neg_a=*/false, av, /*neg_b=*/false, bv,
        /*c_mod=*/(short)0, acc, /*reuse_a=*/false, /*reuse_b=*/false);
  }

  const int n   = nbase + l16;
  const float bn = bias[n];
  const int mofs = (lane < 16) ? 0 : 8;
#pragma unroll
  for (int v = 0; v < 8; ++v) {
    const int m = mbase + v + mofs;
    float y = acc[v] + bn;
    if (MODE == 1) y = fmaxf(y, 0.0f);
    if (MODE == 2) y = extra[m * Ncols + n] * sigmoidf(y);
    out[m * Ncols + n] = y;
  }
}

// ---------------- LayerNorm + ReLU over each [512] row ---------------------
__global__ void ln_relu_kernel(const float* __restrict__ ylin,
                               const float* __restrict__ g,
                               const float* __restrict__ bb,
                               float* __restrict__ outp) {
  const int b = blockIdx.x, mode = blockIdx.y;
  const float* row = ylin + (mode * BB + b) * DD;
  float* orow = outp + (mode * BB + b) * DD;
  const int t = threadIdx.x;
  __shared__ float s[8];
  const float v0 = row[t], v1 = row[t + 256];
  const float mu = block_sum(v0 + v1, s) * (1.0f / 512.0f);
  const float d0 = v0 - mu, d1 = v1 - mu;
  const float var = block_sum(d0 * d0 + d1 * d1, s) * (1.0f / 512.0f);
  const float inv = rsqrtf(var + 1e-5f);
  const float* gm = g + mode * DD;
  const float* bm = bb + mode * DD;
  orow[t]       = fmaxf(d0 * inv * gm[t]       + bm[t],       0.0f);
  orow[t + 256] = fmaxf(d1 * inv * gm[t + 256] + bm[t + 256], 0.0f);
}

// ---------------- per (mode,batch) row sums of wf --------------------------
__global__ void wfsum_kernel(const float* __restrict__ wf,
                             float* __restrict__ wfsum) {
  const int b = blockIdx.x, mode = blockIdx.y;
  const float* row = wf + (mode * BB + b) * DD;
  __shared__ float s[8];
  const float tot = block_sum(row[threadIdx.x] + row[threadIdx.x + 256], s);
  if (threadIdx.x == 0) wfsum[mode * BB + b] = tot;
}

// ---------------- interaction pair kernel ----------------------------------
// One block per (d, mode). PHASE 0: hist==0, accumulates batch-mean into hist.
// PHASE 1: reads hist produced by PHASE 0.  Writes softmax-dot (without the
// modal_bias rank-1 term, added in combine) to att[p][b][d].
template <int PHASE>
__global__ void pair_kernel(const float* __restrict__ wf,
                            const float* __restrict__ constraint,
                            float* __restrict__ hist,
                            float* __restrict__ att) {
  const int d = blockIdx.x;
  const int i = blockIdx.y;
  const int j = (PHASE == 0) ? ((i == 0) ? 1 : 0) : ((i == 2) ? 1 : 2);
  const int p = i * 2 + PHASE;
  const float* wfi = wf + i * BB * DD;
  const float* wfj = wf + j * BB * DD;
  const float c = constraint[i];
  const int t = threadIdx.x;
  const int e0 = t, e1 = t + 256;

  float h0 = 0.0f, h1 = 0.0f;
  if (PHASE == 1) {
    h0 = hist[(i * DD + d) * DD + e0];
    h1 = hist[(i * DD + d) * DD + e1];
  }
  float acc0 = 0.0f, acc1 = 0.0f;
  __shared__ float s[8];

  for (int b = 0; b < BB; ++b) {
    const float sc  = wfi[b * DD + d] * INV_SCALE;
    const float wj0 = wfj[b * DD + e0];
    const float wj1 = wfj[b * DD + e1];
    const float cur0 = sc * wj0, cur1 = sc * wj1;
    const float f0 = SMOOTH * (h0 + fminf(fmaxf(cur0 - h0, -c), c))
                     + (1.0f - SMOOTH) * cur0;
    const float f1 = SMOOTH * (h1 + fminf(fmaxf(cur1 - h1, -c), c))
                     + (1.0f - SMOOTH) * cur1;
    if (PHASE == 0) { acc0 += f0; acc1 += f1; }
    const float m  = block_max(fmaxf(f0, f1), s);
    const float x0 = __expf(f0 - m), x1 = __expf(f1 - m);
    const float rs = block_sum(x0 + x1, s);
    const float dt = block_sum(x0 * wj0 + x1 * wj1, s);
    if (t == 0) att[(p * BB + b) * DD + d] = dt / rs;
  }
  if (PHASE == 0) {
    // hist = (sum_b final0) / (B * HIST) = sum / 320
    hist[(i * DD + d) * DD + e0] = acc0 * (1.0f / 320.0f);
    hist[(i * DD + d) * DD + e1] = acc1 * (1.0f / 320.0f);
  }
}

// ---------------- final combine + attention-weight softmax -----------------
__global__ void combine_kernel(const float* __restrict__ wf,
                               const float* __restrict__ att,
                               const float* __restrict__ wfsum,
                               const float* __restrict__ gamma,
                               const float* __restrict__ mb,
                               float* __restrict__ out) {
  const int t = blockIdx.x * blockDim.x + threadIdx.x;
  if (t < BB * DD) {
    const int b = t >> 9;
    float acc = (wf[t] + wf[BB * DD + t] + wf[2 * BB * DD + t]) * (1.0f / 3.0f);
#pragma unroll
    for (int p = 0; p < 6; ++p) {
      const int i = p >> 1, jj = p & 1;
      const int j = (jj == 0) ? ((i == 0) ? 1 : 0) : ((i == 2) ? 1 : 2);
      const float g = sigmoidf(gamma[i * MM + j]);
      const float a = att[p * BB * DD + t] + mb[i * MM + j] * wfsum[j * BB + b];
      acc += g * a * (1.0f / 6.0f);
    }
    out[t] = acc;
  }
  if (t == 0) {
    float s0, s1, s2;
    {
      float sv[3];
#pragma unroll
      for (int i = 0; i < 3; ++i) {
        const int j0 = (i == 0) ? 1 : 0, j1 = (i == 2) ? 1 : 2;
        sv[i] = 0.5f * (sigmoidf(gamma[i * MM + j0]) + sigmoidf(gamma[i * MM + j1]));
      }
      s0 = sv[0]; s1 = sv[1]; s2 = sv[2];
    }
    const float mx = fmaxf(s0, fmaxf(s1, s2));
    const float e0 = __expf(s0 - mx), e1 = __expf(s1 - mx), e2 = __expf(s2 - mx);
    const float inv = 1.0f / (e0 + e1 + e2);
    out[BB * DD + 0] = e0 * inv;
    out[BB * DD + 1] = e1 * inv;
    out[BB * DD + 2] = e2 * inv;
  }
}

// ---------------------------------------------------------------------------
extern "C" void kernel_launch(void* const* d_in, const int* in_sizes, int n_in,
                              void* d_out, int out_size, void* d_ws, size_t ws_size,
                              hipStream_t stream) {
  const float* x      = (const float*)d_in[0];   // [3,64,512]
  const float* projW  = (const float*)d_in[1];   // [3,512,512]
  const float* projb  = (const float*)d_in[2];   // [3,512]
  const float* lng    = (const float*)d_in[3];   // [3,512]
  const float* lnb    = (const float*)d_in[4];   // [3,512]
  const float* impW1  = (const float*)d_in[5];   // [3,128,512]
  const float* impb1  = (const float*)d_in[6];   // [3,128]
  const float* impW2  = (const float*)d_in[7];   // [3,512,128]
  const float* impb2  = (const float*)d_in[8];   // [3,512]
  const float* gamma  = (const float*)d_in[9];   // [3,3]
  const float* mbias  = (const float*)d_in[10];  // [3,3]
  const float* constr = (const float*)d_in[11];  // [3]
  float* out = (float*)d_out;                    // 64*512 fused + 3 weights
  float* ws  = (float*)d_ws;

  // workspace layout (floats)
  float* proj_lin = ws;                 //  98304  [3][64][512]
  float* proj     = ws + 98304;         //  98304  [3][64][512]
  float* hbuf     = ws + 196608;        //  24576  [3][64][128]
  float* wf       = ws + 221184;        //  98304  [3][64][512]
  float* wfsum    = ws + 319488;        //    192  [3][64]
  float* hist     = ws + 319680;        // 786432  [3][512][512]
  float* att      = ws + 1106112;       // 196608  [6][64][512]

  // 1) projection GEMM (WMMA fp32), 2) LN+ReLU, 3) imp MLP (WMMA), wf
  gemm_wmma<0><<<dim3(32, 4, 3), 32, 0, stream>>>(x, projW, projb, nullptr,
                                                  proj_lin, 512, 512);
  ln_relu_kernel<<<dim3(64, 3), 256, 0, stream>>>(proj_lin, lng, lnb, proj);
  gemm_wmma<1><<<dim3(8, 4, 3), 32, 0, stream>>>(proj, impW1, impb1, nullptr,
                                                 hbuf, 512, 128);
  gemm_wmma<2><<<dim3(32, 4, 3), 32, 0, stream>>>(hbuf, impW2, impb2, proj,
                                                  wf, 128, 512);
  wfsum_kernel<<<dim3(64, 3), 256, 0, stream>>>(wf, wfsum);

  // 4) interaction pairs: phase 0 writes hist, phase 1 consumes it
  pair_kernel<0><<<dim3(512, 3), 256, 0, stream>>>(wf, constr, hist, att);
  pair_kernel<1><<<dim3(512, 3), 256, 0, stream>>>(wf, constr, hist, att);

  // 5) combine + attention-weight softmax
  combine_kernel<<<dim3(128), 256, 0, stream>>>(wf, att, wfsum, gamma, mbias, out);
}